// GAT_22393959481367
// MI455X (gfx1250) — compile-verified
//
#include <hip/hip_runtime.h>
#include <math.h>

#define B_ 8
#define N_ 2048
#define FIN 128
#define FOUT 64
#define ALPHA 0.2f
#define LOG2E 1.4426950408889634f
#define MASK_PITCH 2052   // 2048 + 4 bytes pad -> avoids 16-way LDS bank conflicts

typedef float v2f __attribute__((ext_vector_type(2)));
typedef float v4f __attribute__((ext_vector_type(4)));
typedef float v8f __attribute__((ext_vector_type(8)));

__device__ __forceinline__ float wave_max_f(float v) {
#pragma unroll
  for (int o = 16; o > 0; o >>= 1) v = fmaxf(v, __shfl_xor(v, o, 32));
  return v;
}
__device__ __forceinline__ float wave_sum_f(float v) {
#pragma unroll
  for (int o = 16; o > 0; o >>= 1) v += __shfl_xor(v, o, 32);
  return v;
}

// exp(x) for x <= 0 via a single v_exp_f32 (no range fixup needed)
__device__ __forceinline__ float fast_exp_neg(float x) {
  return __builtin_amdgcn_exp2f(x * LOG2E);
}

// ---------------------------------------------------------------------------
// Kernel 1: h[b*N+n][o] = sum_k afms[b,n,k] * W[k,o]   (0.27 GFLOP, trivial)
// block = 256 threads = 4 rows x 64 output features
// ---------------------------------------------------------------------------
__global__ __launch_bounds__(256) void gat_h_kernel(const float* __restrict__ afms,
                                                    const float* __restrict__ W,
                                                    float* __restrict__ h) {
  int tid = threadIdx.x;
  int row = blockIdx.x * 4 + (tid >> 6);
  int o = tid & 63;
  const float* ar = afms + (size_t)row * FIN;
  float acc = 0.f;
#pragma unroll 8
  for (int k = 0; k < FIN; ++k) acc = fmaf(ar[k], W[k * FOUT + o], acc);
  h[(size_t)row * FOUT + o] = acc;
}

// ---------------------------------------------------------------------------
// Kernel 2: s1[r] = h[r,:].a1 ; s2[r] = h[r,:].a2   (one wave32 per row)
// ---------------------------------------------------------------------------
__global__ __launch_bounds__(256) void gat_s_kernel(const float* __restrict__ h,
                                                    const float* __restrict__ a1,
                                                    const float* __restrict__ a2,
                                                    float* __restrict__ s1,
                                                    float* __restrict__ s2) {
  int tid = threadIdx.x, w = tid >> 5, lane = tid & 31;
  int row = blockIdx.x * 8 + w;
  const float* hr = h + (size_t)row * FOUT;
  float h0 = hr[lane], h1 = hr[lane + 32];
  float p1 = h0 * a1[lane] + h1 * a1[lane + 32];
  float p2 = h0 * a2[lane] + h1 * a2[lane + 32];
  p1 = wave_sum_f(p1);
  p2 = wave_sum_f(p2);
  if (lane == 0) { s1[row] = p1; s2[row] = p2; }
}

// ---------------------------------------------------------------------------
// Kernel 3: fused  A-output + mask + masked row softmax + (att @ h) via
//           V_WMMA_F32_16X16X4_F32, + ReLU.
// grid = (N/16, B), block = 256 threads (8 wave32).  Each block owns a
// 16-row tile of one batch.
// ---------------------------------------------------------------------------
__global__ __launch_bounds__(256) void gat_main_kernel(
    const float* __restrict__ adjs, const float* __restrict__ h,
    const float* __restrict__ s1g, const float* __restrict__ s2g,
    float* __restrict__ x_out, float* __restrict__ A_out) {
  // sm_pool is aliased: bytes [mask: 16*2052 = 32832 B] during phases 1-3,
  // then [partials: 8*16*64 floats = 32768 B] after the post-K-loop barrier.
  __shared__ float sm_pool[8208];          // 32832 bytes
  __shared__ float sm_s2[N_];              // 8192 bytes
  __shared__ float sm_s1[16];
  __shared__ float sm_m[16];
  __shared__ float sm_iz[16];

  unsigned char* sm_mask = (unsigned char*)sm_pool;
  float* sm_part = sm_pool;

  int tid = threadIdx.x, w = tid >> 5, lane = tid & 31;
  int b = blockIdx.y;
  int i0 = blockIdx.x * 16;
  size_t bN = (size_t)b * N_;

  // ---- stage s2 row (2048) and s1 for this 16-row tile --------------------
  for (int j = tid; j < N_; j += 256) sm_s2[j] = s2g[bN + j];
  if (tid < 16) sm_s1[tid] = s1g[bN + i0 + tid];

  // ---- Phase 1: stream adjs once -> A, build mask, fix diagonal -----------
  for (int r = 0; r < 2; ++r) {
    int il = 2 * w + r;
    int ig = i0 + il;
    const float* arow = adjs + (bN + ig) * N_;
    float* Arow = A_out + (bN + ig) * N_;
    unsigned char* mrow = sm_mask + il * MASK_PITCH;
    float amax = 0.f;
    for (int j4 = lane * 4; j4 < N_; j4 += 128) {
      v4f v = __builtin_nontemporal_load(reinterpret_cast<const v4f*>(arow + j4));
      amax = fmaxf(amax, fmaxf(fmaxf(v.x, v.y), fmaxf(v.z, v.w)));
      __builtin_nontemporal_store(v, reinterpret_cast<v4f*>(Arow + j4));
      mrow[j4 + 0] = v.x > 0.f;
      mrow[j4 + 1] = v.y > 0.f;
      mrow[j4 + 2] = v.z > 0.f;
      mrow[j4 + 3] = v.w > 0.f;
    }
    amax = wave_max_f(amax);                 // mask_blank for this row
    if (lane == 0) {
      float diag = arow[ig] + amax;          // A = adjs + mask_blank * I
      Arow[ig] = diag;
      mrow[ig] = diag > 0.f;
    }
  }
  __syncthreads();

  // ---- Phase 2: exact masked softmax stats (max then sum-exp) per row -----
  for (int r = 0; r < 2; ++r) {
    int il = 2 * w + r;
    const unsigned char* mrow = sm_mask + il * MASK_PITCH;
    float s1v = sm_s1[il];
    float m = -3.0e38f;
    for (int j = lane; j < N_; j += 32) {
      if (mrow[j]) {
        float t = s1v + sm_s2[j];
        float e = t > 0.f ? t : ALPHA * t;
        m = fmaxf(m, e);
      }
    }
    m = wave_max_f(m);
    float Z = 0.f;
    for (int j = lane; j < N_; j += 32) {
      if (mrow[j]) {
        float t = s1v + sm_s2[j];
        float e = t > 0.f ? t : ALPHA * t;
        Z += fast_exp_neg(e - m);            // e - m <= 0 always
      }
    }
    Z = wave_sum_f(Z);
    if (lane == 0) {
      sm_m[il] = m;
      sm_iz[il] = (Z > 0.f) ? (1.f / Z) : 0.f;   // fully-masked row -> 0
    }
  }
  __syncthreads();

  // ---- Phase 3: h_prime = att @ h with V_WMMA_F32_16X16X4_F32 -------------
  // Wave w owns K-range [w*256, w*256+256). 16x64 accumulator = 4 N-tiles.
  // A frag (16x4 f32): lanes hold M = lane&15; lane-half selects K 0-1 / 2-3.
  // B frag (4x16 f32): lanes hold N = lane&15; same K split.
  int half = lane >> 4;
  int col = lane & 15;
  int koff = half << 1;
  float s1r = sm_s1[col];
  float mr = sm_m[col];
  float izr = sm_iz[col];
  const unsigned char* mrowA = sm_mask + col * MASK_PITCH;
  const float* hb = h + bN * FOUT;

  v8f c0 = {}, c1 = {}, c2 = {}, c3 = {};
  int jbeg = w * 256, jend = jbeg + 256;
  for (int j0 = jbeg; j0 < jend; j0 += 4) {
    int ja = j0 + koff;
    // synthesize att values for this lane's two (M=col, K=ja/ja+1) entries
    float t0 = s1r + sm_s2[ja];
    float e0 = t0 > 0.f ? t0 : ALPHA * t0;
    float v0 = fast_exp_neg(e0 - mr) * izr;
    float att0 = mrowA[ja] ? v0 : 0.f;
    float t1 = s1r + sm_s2[ja + 1];
    float e1 = t1 > 0.f ? t1 : ALPHA * t1;
    float v1 = fast_exp_neg(e1 - mr) * izr;
    float att1 = mrowA[ja + 1] ? v1 : 0.f;
    v2f a = {att0, att1};

    const float* hr0 = hb + (size_t)ja * FOUT + col;   // K row ja
    const float* hr1 = hr0 + FOUT;                     // K row ja+1
    v2f b0 = {hr0[0],  hr1[0]};
    v2f b1 = {hr0[16], hr1[16]};
    v2f b2 = {hr0[32], hr1[32]};
    v2f b3 = {hr0[48], hr1[48]};

    c0 = __builtin_amdgcn_wmma_f32_16x16x4_f32(false, a, false, b0, (short)0, c0, false, false);
    c1 = __builtin_amdgcn_wmma_f32_16x16x4_f32(false, a, false, b1, (short)0, c1, false, false);
    c2 = __builtin_amdgcn_wmma_f32_16x16x4_f32(false, a, false, b2, (short)0, c2, false, false);
    c3 = __builtin_amdgcn_wmma_f32_16x16x4_f32(false, a, false, b3, (short)0, c3, false, false);
  }

  __syncthreads();   // all mask reads done -> safe to alias sm_pool as partials

  // C layout: VGPR v -> M = v + half*8, N = tile_n0 + (lane&15)
  float* part = sm_part + w * 1024;
  int rowc_base = half * 8;
#pragma unroll
  for (int v = 0; v < 8; ++v) {
    int rowc = rowc_base + v;
    part[rowc * 64 + 0  + col] = c0[v];
    part[rowc * 64 + 16 + col] = c1[v];
    part[rowc * 64 + 32 + col] = c2[v];
    part[rowc * 64 + 48 + col] = c3[v];
  }
  __syncthreads();

  // reduce 8 wave-partials, ReLU, store x
  for (int idx = tid; idx < 1024; idx += 256) {
    float sum = 0.f;
#pragma unroll
    for (int ww = 0; ww < 8; ++ww) sum += sm_part[ww * 1024 + idx];
    int il = idx >> 6, o = idx & 63;
    x_out[(bN + i0 + il) * FOUT + o] = fmaxf(sum, 0.f);
  }
}

// ---------------------------------------------------------------------------
extern "C" void kernel_launch(void* const* d_in, const int* in_sizes, int n_in,
                              void* d_out, int out_size, void* d_ws, size_t ws_size,
                              hipStream_t stream) {
  (void)in_sizes; (void)n_in; (void)out_size; (void)ws_size;
  const float* adjs = (const float*)d_in[0];
  const float* afms = (const float*)d_in[1];
  // d_in[2..6] = Type/Order/Arom/Conj/RingAtt: unused by reference
  const float* W  = (const float*)d_in[7];
  const float* a1 = (const float*)d_in[8];
  const float* a2 = (const float*)d_in[9];

  float* out = (float*)d_out;
  float* x_out = out;                                    // [B,N,64]
  float* A_out = out + (size_t)B_ * N_ * FOUT;           // [B,N,N]

  float* h  = (float*)d_ws;                              // [B*N,64] = 4 MB
  float* s1 = h + (size_t)B_ * N_ * FOUT;                // [B*N]
  float* s2 = s1 + (size_t)B_ * N_;                      // [B*N]

  gat_h_kernel<<<(B_ * N_) / 4, 256, 0, stream>>>(afms, W, h);
  gat_s_kernel<<<(B_ * N_) / 8, 256, 0, stream>>>(h, a1, a2, s1, s2);
  dim3 grid(N_ / 16, B_);
  gat_main_kernel<<<grid, 256, 0, stream>>>(adjs, h, s1, s2, x_out, A_out);
}